// FNRDNet_23742579212333
// MI455X (gfx1250) — compile-verified
//
#include <hip/hip_runtime.h>

// ---------------------------------------------------------------------------
// Fused classifier head + outlier count for MI455X (gfx1250).
//   pred[512,10] = act[512,180736] @ w[180736,10]          (f32 WMMA 16x16x4)
//   cfd[512]     = 1 - (#outliers per row)/180736
// Bandwidth-bound: 370 MB of activations streamed once (~16 us @ 23.3 TB/s).
// v4: activation async-DMA loads marked non-temporal (th:TH_LOAD_NT) so the
//     one-shot 370 MB stream doesn't evict the L2-resident w/mask operands.
//     Otherwise identical to v3: async double-buffered LDS staging, one
//     barrier per chunk, LDS-padded B operand, WMMA f32 16x16x4 accumulate,
//     exact integer outlier counts, LDS + global atomic reductions.
// ---------------------------------------------------------------------------

#define MASK_SIZE   180736
#define NUM_CLASSES 10
#define BATCH       512

#define KBLOCKS          353   // gridDim.x ; 353*512 == 180736 exactly
#define ROWTILES         32    // gridDim.y ; 32*16   == 512    exactly
#define CHUNK_K          128   // columns staged per LDS tile
#define CHUNKS_PER_BLOCK 4     // 4*128 = 512 K-columns per block
#define BLOCK_K          (CHUNK_K * CHUNKS_PER_BLOCK)
#define LDSW             132   // padded LDS row stride (floats) -> conflict-free

typedef __attribute__((ext_vector_type(2))) float v2f;
typedef __attribute__((ext_vector_type(8))) float v8f;

// -------------------- init: zero all of d_out (pred + count region) --------
__global__ void fnrd_init(float* __restrict__ out, int n) {
    int i = blockIdx.x * 256 + threadIdx.x;
    if (i < n) out[i] = 0.0f;
}

// async copy of one b128 (4 floats) from global memory directly into LDS.
// LDS byte offset = low 32 bits of the generic shared-memory pointer
// (flat LDS aperture keeps the DS offset in addr[31:0]).
// Non-temporal: activations are streamed exactly once; don't pollute L2.
__device__ __forceinline__ void async_b128_to_lds_nt(const float* gptr, const float* lptr) {
    const unsigned           ldsoff = (unsigned)(unsigned long long)lptr;
    const unsigned long long gaddr  = (unsigned long long)gptr;
    asm volatile("global_load_async_to_lds_b128 %0, %1, off th:TH_LOAD_NT"
                 :: "v"(ldsoff), "v"(gaddr) : "memory");
}

__device__ __forceinline__ void wait_async0() {
    asm volatile("s_wait_asynccnt 0x0" ::: "memory");
}

// -------------------- main fused kernel ------------------------------------
__global__ __launch_bounds__(256) void fnrd_main(
    const float* __restrict__ act,   // [512, 180736] row-major
    const float* __restrict__ wmat,  // [180736, 10]  row-major
    const float* __restrict__ mxm,   // [180736]
    const float* __restrict__ mnm,   // [180736]
    float* __restrict__ pred_out,    // [512*10]  (atomically accumulated)
    float* __restrict__ cnt_out)     // [512]     (integer-valued f32, exact)
{
    __shared__ float tiles[2][16 * LDSW];   // double-buffered 16x128 act tiles
    __shared__ float wlds[BLOCK_K * 16];    // w slice, zero-padded to 16 cols
    __shared__ float predtile[256];         // 16 x 16 cross-wave reduction tile
    __shared__ unsigned int cnttile[16];    // per-row outlier counts

    const int t    = threadIdx.x;
    const int lane = t & 31;                // wave32
    const int wave = t >> 5;                // 8 waves
    const int rowbase = blockIdx.y * 16;
    const int kbase   = blockIdx.x * BLOCK_K;

    predtile[t] = 0.0f;
    if (t < 16) cnttile[t] = 0u;

    // staging map: thread t owns float4 slots {t, t+256} of the 16x128 tile
    const int r0 = t >> 5,         c0 = (t & 31) * 4;
    const int r1 = (t + 256) >> 5, c1 = (t & 31) * 4;   // row + 8, same col
    const float* grow0 = &act[(size_t)(rowbase + r0) * MASK_SIZE + kbase];
    const float* grow1 = &act[(size_t)(rowbase + r1) * MASK_SIZE + kbase];

    // prime the pipeline: async-stage chunk 0 into buffer 0
    async_b128_to_lds_nt(grow0, &tiles[0][r0 * LDSW + c0]);
    async_b128_to_lds_nt(grow1, &tiles[0][r1 * LDSW + c1]);

    // ---- stage w slice to LDS (padded [512][16]) while the DMA runs ------
    {
        const float* wsrc = wmat + (size_t)kbase * NUM_CLASSES;
        // valid entries: 5120 floats as 2560 float2 pairs (never straddle a
        // k-row: rows are 10 floats, pairs start at even offsets 0,2,4,6,8)
#pragma unroll
        for (int it = 0; it < 10; ++it) {
            const int p    = t + it * 256;       // pair index 0..2559
            const int flat = p * 2;              // 0..5118, even
            const int k    = flat / NUM_CLASSES; // 0..511
            const int n    = flat - k * NUM_CLASSES;  // 0,2,4,6,8
            const float2 v = *(const float2*)&wsrc[flat];
            *(float2*)&wlds[k * 16 + n] = v;     // ds_store_b64, 8B aligned
        }
        // zero padding: columns 10..15, 512 k x 3 pairs = 1536 b64 stores
#pragma unroll
        for (int it = 0; it < 6; ++it) {
            const int s = t + it * 256;          // 0..1535
            const int k = s & 511;
            const int n = 10 + 2 * (s >> 9);     // 10, 12, 14
            const float2 z = make_float2(0.0f, 0.0f);
            *(float2*)&wlds[k * 16 + n] = z;
        }
    }

    // WMMA 16x16x4 f32 operand layout (wave32):
    //   A (16x4): lanes 0-15 -> M=lane, {K0,K1}; lanes 16-31 -> M=lane-16, {K2,K3}
    //   B (4x16): lanes 0-15 -> N=lane, {K0,K1}; lanes 16-31 -> N=lane-16, {K2,K3}
    const int M    = lane & 15;
    const int hf   = lane >> 4;       // 0 or 1
    const int koff = hf * 2;          // this lane's K offset within a 4-group
    const int N    = lane & 15;

    v8f acc = {};                     // per-wave 16x16 f32 accumulator
    unsigned int cnt = 0u;            // per-lane outlier count (row M of tile)

    for (int c = 0; c < CHUNKS_PER_BLOCK; ++c) {
        wait_async0();        // my chunk-c transfers have landed in LDS
        __syncthreads();      // everyone's chunk c (and wlds) ready

        if (c + 1 < CHUNKS_PER_BLOCK) {     // overlap: DMA chunk c+1 now
            const int kn = (c + 1) * CHUNK_K;
            float* buf = tiles[(c + 1) & 1];
            async_b128_to_lds_nt(grow0 + kn, &buf[r0 * LDSW + c0]);
            async_b128_to_lds_nt(grow1 + kn, &buf[r1 * LDSW + c1]);
        }

        const float* tile = tiles[c & 1];
        const int kc0 = kbase + c * CHUNK_K;

        // ---- wave `wave` consumes local K columns [16*wave, 16*wave+16) ----
#pragma unroll
        for (int j = 0; j < 4; ++j) {
            const int kk   = wave * 16 + j * 4;       // local 4-group base
            const int kl   = c * CHUNK_K + kk + koff; // block-local k of a.x
            const int kcol = kc0 + kk + koff;         // global column of a.x

            const v2f a = *(const v2f*)&tile[M * LDSW + kk + koff]; // ds_load_b64

            // fused outlier count: each (row,col) element hit exactly once
            const float mx0 = mxm[kcol],     mn0 = mnm[kcol];
            const float mx1 = mxm[kcol + 1], mn1 = mnm[kcol + 1];
            cnt += (unsigned)((a.x > mx0) + (a.x < mn0) + (a.y > mx1) + (a.y < mn1));

            // B operand: padded w from LDS, unconditional ds_load_b32 x2
            v2f b;
            b.x = wlds[kl * 16 + N];
            b.y = wlds[(kl + 1) * 16 + N];

            acc = __builtin_amdgcn_wmma_f32_16x16x4_f32(
                false, a, false, b, (short)0, acc, false, false);
        }
    }

    // ---- cross-wave reduction of the 16x16 pred tile (ds_add_f32) ----
    // D layout: lane L, VGPR j -> element (M = j + 8*(L>>4), N = L&15)
#pragma unroll
    for (int j = 0; j < 8; ++j) {
        const int m = j + 8 * hf;
        atomicAdd(&predtile[m * 16 + N], acc[j]);
    }
    atomicAdd(&cnttile[M], cnt);
    __syncthreads();

    // ---- global accumulation ----
    if (t < 16) atomicAdd(&cnt_out[rowbase + t], (float)cnttile[t]); // exact int-f32
    {
        const int m = t >> 4, n = t & 15;
        if (n < NUM_CLASSES)
            atomicAdd(&pred_out[(size_t)(rowbase + m) * NUM_CLASSES + n], predtile[t]);
    }
}

// -------------------- finalize: counts -> cfd, in place --------------------
__global__ void fnrd_finalize(float* __restrict__ cfd) {
    int i = blockIdx.x * 256 + threadIdx.x;
    if (i < BATCH) {
        float c = cfd[i];                       // integer-valued count
        cfd[i] = 1.0f - c / (float)MASK_SIZE;
    }
}

// ---------------------------------------------------------------------------
extern "C" void kernel_launch(void* const* d_in, const int* in_sizes, int n_in,
                              void* d_out, int out_size, void* d_ws, size_t ws_size,
                              hipStream_t stream) {
    (void)in_sizes; (void)n_in; (void)d_ws; (void)ws_size; (void)out_size;

    const float* act = (const float*)d_in[0];
    const float* w   = (const float*)d_in[1];
    const float* mxm = (const float*)d_in[2];
    const float* mnm = (const float*)d_in[3];

    float* pred = (float*)d_out;                    // [512*10]
    float* cfd  = pred + BATCH * NUM_CLASSES;       // [512] (count acc, then cfd)

    const int total = BATCH * NUM_CLASSES + BATCH;  // 5632
    fnrd_init<<<(total + 255) / 256, 256, 0, stream>>>(pred, total);

    dim3 grid(KBLOCKS, ROWTILES);
    fnrd_main<<<grid, 256, 0, stream>>>(act, w, mxm, mnm, pred, cfd);

    fnrd_finalize<<<(BATCH + 255) / 256, 256, 0, stream>>>(cfd);
}